// InteractionBlock_17815524344047
// MI455X (gfx1250) — compile-verified
//
#include <hip/hip_runtime.h>

// MACE-style interaction block for MI455X (gfx1250), fp32 WMMA path.
// ws layout (floats): xs0[10000*64] | xs1[10000*192] | agg[10000*1344]  => 64 MB total.

#define N_NODES 10000
#define N_EDGES 160000

typedef float v2f __attribute__((ext_vector_type(2)));
typedef float v4f __attribute__((ext_vector_type(4)));
typedef float v8f __attribute__((ext_vector_type(8)));

__device__ __forceinline__ v8f wmma_f32_4(v2f a, v2f b, v8f c) {
  // D(16x16,f32) += A(16x4,f32) * B(4x16,f32); wave32, f32 accumulate
  return __builtin_amdgcn_wmma_f32_16x16x4_f32(false, a, false, b, (short)0, c,
                                               false, false);
}

// silu via hardware v_rcp_f32 (1 ulp) instead of the IEEE div expansion.
__device__ __forceinline__ float silu_f(float x) {
  return x * __builtin_amdgcn_rcpf(1.0f + __expf(-x));
}

__device__ __forceinline__ v2f ld2(const float* p) {   // 8B-aligned pair load
  return *reinterpret_cast<const v2f*>(p);
}

// Generic pointers to LDS carry the LDS byte offset in addr[31:0]
// (per the ISA aperture rule); truncate to get VDST offset.
__device__ __forceinline__ unsigned lds_off(const void* p) {
  return (unsigned)(uintptr_t)p;
}

// Async copy 16B from global to LDS (per active lane); tracked by ASYNCcnt.
__device__ __forceinline__ void async_ld_b128(unsigned ldsoff, const float* g) {
  asm volatile("global_load_async_to_lds_b128 %0, %1, off"
               :: "v"(ldsoff), "v"(g) : "memory");
}

__device__ __forceinline__ void wait_async0() {
  asm volatile("s_wait_asynccnt 0x0" ::: "memory");
}

__global__ void zero_kernel4(v4f* __restrict__ p, int n4) {
  int i = blockIdx.x * blockDim.x + threadIdx.x;
  if (i < n4) p[i] = (v4f){0.f, 0.f, 0.f, 0.f};
}

// ---------------------------------------------------------------------------
// Kernel 1: node channel linears (templated on layout so A-strides fold into
// load immediates).
//   X1=false: xs0[n][v]       = (sum_u nf[n][u]        * Wl0[u][v]) / 8
//   X1=true : xs1[n][v*3 + i] = (sum_u nf[n][64+u*3+i] * Wl1[u][v]) / 8
// grid = (625 node tiles, 1 or 3 components), block = 128 (4 waves)
// ---------------------------------------------------------------------------
template <bool X1>
__global__ void __launch_bounds__(128)
node_lin_t(const float* __restrict__ nf, const float* __restrict__ W,
           float* __restrict__ dst)
{
  const int tile = blockIdx.x;
  const int i    = X1 ? (int)blockIdx.y : 0;   // x1 component
  const int wid  = threadIdx.x >> 5;
  const int lane = threadIdx.x & 31;
  const int h    = lane >> 4;
  const int lr   = lane & 15;
  const int v0   = wid * 16;
  const int n    = tile * 16 + lr;             // A-row for this lane

  constexpr int ASTR = X1 ? 3 : 1;
  const float* Abase = nf + (size_t)n * 256 + (X1 ? (64 + i) : 0) + 2 * h * ASTR;
  const float* Wb    = W + v0 + lr + 2 * h * 64;

  v8f c = {};
#pragma unroll
  for (int k0 = 0; k0 < 64; k0 += 4) {
    v2f a, b;
    a.x = Abase[k0 * ASTR];
    a.y = Abase[(k0 + 1) * ASTR];
    b.x = Wb[k0 * 64];
    b.y = Wb[(k0 + 1) * 64];
    c = wmma_f32_4(a, b, c);
  }
#pragma unroll
  for (int j = 0; j < 8; ++j) {
    const int row = tile * 16 + j + 8 * h;
    const int col = v0 + lr;
    const float val = c[j] * 0.125f;           // 1/sqrt(64)
    if (X1) dst[(size_t)row * 192 + col * 3 + i] = val;
    else    dst[(size_t)row * 64 + col] = val;
  }
}

// ---------------------------------------------------------------------------
// Kernel 2: fused edge MLP (WMMA) + CG tensor product + atomic scatter.
// Wm1/Wm2/Wm3 staged into LDS once per block via async-to-LDS DMA; h kept in
// per-wave LDS ping-pong buffers; each 16-col tile of r (448 cols) consumed
// straight from the WMMA accumulator. block = 128 (4 waves), grid = 2500.
// ---------------------------------------------------------------------------
__global__ void __launch_bounds__(128)
edge_kernel(const float* __restrict__ ea, const float* __restrict__ ef,
            const int* __restrict__ eidx,
            const float* __restrict__ Wm1, const float* __restrict__ Wm2,
            const float* __restrict__ Wm3, const float* __restrict__ Wm4,
            const float* __restrict__ xs0, const float* __restrict__ xs1,
            float* __restrict__ agg)
{
  __shared__ float hb[4][2][16][64];   // per-wave ping-pong hidden buffers (32KB)
  __shared__ float wm1s[8 * 64];       // staged MLP weights (34.8KB total)
  __shared__ float wm2s[64 * 64];
  __shared__ float wm3s[64 * 64];
  __shared__ float shs[4][16][9];      // spherical harmonics per edge
  __shared__ int   sndS[4][16];
  __shared__ int   rcvS[4][16];

  const int tid  = threadIdx.x;
  const int wid  = tid >> 5;
  const int lane = tid & 31;
  const int h    = lane >> 4;
  const int lr   = lane & 15;
  const int et   = blockIdx.x * 4 + wid;
  const int e0   = et * 16;

  // ---- async DMA: stage Wm1 (512f) + Wm2/Wm3 (4096f each) into LDS
  async_ld_b128(lds_off(&wm1s[tid * 4]), Wm1 + tid * 4);      // 128 x 16B
#pragma unroll
  for (int i = 0; i < 8; ++i) {
    const int q = (i * 128 + tid) * 4;
    async_ld_b128(lds_off(&wm2s[q]), Wm2 + q);
    async_ld_b128(lds_off(&wm3s[q]), Wm3 + q);
  }

  if (lane < 16) {
    sndS[wid][lane] = eidx[e0 + lane];
    rcvS[wid][lane] = eidx[N_EDGES + e0 + lane];
  }
  for (int t = lane; t < 16 * 9; t += 32)
    shs[wid][t / 9][t % 9] = ea[(size_t)(e0 + t / 9) * 9 + (t % 9)];

  wait_async0();          // ASYNCcnt == 0: weight tiles landed in LDS
  __syncthreads();

  // -- layer 1: h1 = silu(ef(16x8) @ Wm1(8x64) / sqrt(8)) -> hb[..][0]
  for (int ct = 0; ct < 4; ++ct) {
    v8f c = {};
    const float* wb = &wm1s[ct * 16 + lr + 2 * h * 64];
    const float* ar = &ef[(size_t)(e0 + lr) * 8 + 2 * h];
#pragma unroll
    for (int k0 = 0; k0 < 8; k0 += 4) {
      v2f a = ld2(&ar[k0]);
      v2f b;
      b.x = wb[k0 * 64];
      b.y = wb[(k0 + 1) * 64];
      c = wmma_f32_4(a, b, c);
    }
#pragma unroll
    for (int j = 0; j < 8; ++j)
      hb[wid][0][j + 8 * h][ct * 16 + lr] = silu_f(c[j] * 0.35355339059f);
  }
  __syncthreads();

  // -- layers 2 & 3: h = silu(h @ W / 8); buffers 0->1 then 1->0
  for (int layer = 0; layer < 2; ++layer) {
    const float* Wl = (layer == 0) ? wm2s : wm3s;
    const int src = layer, dst = 1 - layer;
    for (int ct = 0; ct < 4; ++ct) {
      v8f c = {};
      const float* wb = &Wl[ct * 16 + lr + 2 * h * 64];
      const float* ar = &hb[wid][src][lr][2 * h];
#pragma unroll
      for (int k0 = 0; k0 < 64; k0 += 4) {
        v2f a = ld2(&ar[k0]);
        v2f b;
        b.x = wb[k0 * 64];
        b.y = wb[(k0 + 1) * 64];
        c = wmma_f32_4(a, b, c);
      }
#pragma unroll
      for (int j = 0; j < 8; ++j)
        hb[wid][dst][j + 8 * h][ct * 16 + lr] = silu_f(c[j] * 0.125f);
    }
    __syncthreads();
  }

  // -- layer 4 tiles (r = h @ Wm4 / 8) consumed directly: tensor product + scatter
  const float is2 = 0.70710678119f;  // 1/sqrt(2)
  const float is3 = 0.57735026919f;  // 1/sqrt(3)
  const float is5 = 0.44721359550f;  // 1/sqrt(5)
  const float is6 = 0.40824829046f;  // 1/sqrt(6)

  for (int ct = 0; ct < 28; ++ct) {
    v8f c = {};
    const float* wb = Wm4 + ct * 16 + lr + 2 * h * 448;
    const float* ar = &hb[wid][0][lr][2 * h];
#pragma unroll
    for (int k0 = 0; k0 < 64; k0 += 4) {
      v2f a = ld2(&ar[k0]);
      v2f b;
      b.x = wb[k0 * 448];
      b.y = wb[(k0 + 1) * 448];
      c = wmma_f32_4(a, b, c);
    }
    const int p = ct >> 2;                 // path id: uniform across the wave
    const int u = (ct & 3) * 16 + lr;      // channel within path
#pragma unroll 1
    for (int j = 0; j < 8; ++j) {
      const int   el  = j + 8 * h;         // edge within tile
      const float rv  = c[j] * 0.125f;     // r[e][p*64+u]
      const int   s   = sndS[wid][el];
      float*      agr = agg + (size_t)rcvS[wid][el] * 1344;
      const float* sv = shs[wid][el];
      const float* xr = xs1 + (size_t)s * 192 + u * 3;
      if (p == 0) {                                  // (0,0,0)
        atomicAdd(&agr[u], xs0[(size_t)s * 64 + u] * sv[0] * rv);
      } else if (p == 1) {                           // (0,1,1)
        const float f = xs0[(size_t)s * 64 + u] * rv * is3;
        atomicAdd(&agr[64 + u * 3 + 0], f * sv[1]);
        atomicAdd(&agr[64 + u * 3 + 1], f * sv[2]);
        atomicAdd(&agr[64 + u * 3 + 2], f * sv[3]);
      } else if (p == 2) {                           // (0,2,2)
        const float f = xs0[(size_t)s * 64 + u] * rv * is5;
        for (int k = 0; k < 5; ++k)
          atomicAdd(&agr[256 + u * 5 + k], f * sv[4 + k]);
      } else if (p == 3) {                           // (1,0,1)
        const float f = sv[0] * rv * is3;
        for (int k = 0; k < 3; ++k)
          atomicAdd(&agr[576 + u * 3 + k], f * xr[k]);
      } else if (p == 4) {                           // (1,1,0)
        atomicAdd(&agr[768 + u],
                  (xr[0] * sv[1] + xr[1] * sv[2] + xr[2] * sv[3]) * rv * is3);
      } else if (p == 5) {                           // (1,1,2)
        const float xa = xr[0], xb = xr[1], xc = xr[2];
        const float sa = sv[1], sb = sv[2], sc = sv[3];
        const float g  = rv * is5;
        atomicAdd(&agr[832 + u * 5 + 0], (xa * sb + xb * sa) * is2 * g);
        atomicAdd(&agr[832 + u * 5 + 1], (xa * sc + xc * sa) * is2 * g);
        atomicAdd(&agr[832 + u * 5 + 2], (xb * sc + xc * sb) * is2 * g);
        atomicAdd(&agr[832 + u * 5 + 3], (xa * sa - xb * sb) * is2 * g);
        atomicAdd(&agr[832 + u * 5 + 4],
                  (2.0f * xc * sc - xa * sa - xb * sb) * is6 * g);
      } else {                                       // (1,2,1)
        const float xa = xr[0], xb = xr[1], xc = xr[2];
        const float g  = rv * is5;
        const float o0 = (sv[4] * xb + sv[5] * xc + sv[7] * xa) * is2 - sv[8] * xa * is6;
        const float o1 = (sv[4] * xa + sv[6] * xc - sv[7] * xb) * is2 - sv[8] * xb * is6;
        const float o2 = (sv[5] * xa + sv[6] * xb) * is2 + 2.0f * sv[8] * xc * is6;
        atomicAdd(&agr[1152 + u * 3 + 0], o0 * g);
        atomicAdd(&agr[1152 + u * 3 + 1], o1 * g);
        atomicAdd(&agr[1152 + u * 3 + 2], o2 * g);
      }
    }
  }
}

// ---------------------------------------------------------------------------
// Kernel 3: output linears, templated on the (compile-time) gather stride so
// the piecewise-affine agg gather folds into load immediates.
// out[n][o][cmp0 + blockIdx.y]; segment bases passed at launch.
// ---------------------------------------------------------------------------
template <int STRIDE, int NSEG>
__global__ void __launch_bounds__(128)
out_lin_t(const float* __restrict__ agg, const float* __restrict__ W,
          float* __restrict__ out, int cmp0, int b0, int b1, int b2, float scale)
{
  const int tile = blockIdx.x;
  const int k    = blockIdx.y;          // component within this group
  const int cmp  = cmp0 + k;
  const int wid  = threadIdx.x >> 5;
  const int lane = threadIdx.x & 31;
  const int h    = lane >> 4;
  const int lr   = lane & 15;
  const int o0   = wid * 16;

  int seg[3] = {b0 + k, b1 + k, b2 + k};
  const int n = tile * 16 + lr;

  v8f c = {};
#pragma unroll
  for (int sgi = 0; sgi < NSEG; ++sgi) {
    const float* arow = agg + (size_t)n * 1344 + seg[sgi] + 2 * h * STRIDE;
    const float* wb   = W + (size_t)sgi * 64 * 64 + o0 + lr + 2 * h * 64;
#pragma unroll
    for (int k0 = 0; k0 < 64; k0 += 4) {
      v2f a, b;
      a.x = arow[k0 * STRIDE];
      a.y = arow[(k0 + 1) * STRIDE];
      b.x = wb[k0 * 64];
      b.y = wb[(k0 + 1) * 64];
      c = wmma_f32_4(a, b, c);
    }
  }
#pragma unroll
  for (int j = 0; j < 8; ++j) {
    const int row = tile * 16 + j + 8 * h;
    out[(size_t)row * 576 + (o0 + lr) * 9 + cmp] = c[j] * scale;
  }
}

// ---------------------------------------------------------------------------
extern "C" void kernel_launch(void* const* d_in, const int* in_sizes, int n_in,
                              void* d_out, int out_size, void* d_ws, size_t ws_size,
                              hipStream_t stream)
{
  const float* nf  = (const float*)d_in[0];
  // d_in[1] = node_attrs : unused by the reference
  const float* ea  = (const float*)d_in[2];
  const float* ef  = (const float*)d_in[3];
  const int*   ei  = (const int*)  d_in[4];
  const float* Wl0 = (const float*)d_in[5];
  const float* Wl1 = (const float*)d_in[6];
  const float* Wm1 = (const float*)d_in[7];
  const float* Wm2 = (const float*)d_in[8];
  const float* Wm3 = (const float*)d_in[9];
  const float* Wm4 = (const float*)d_in[10];
  const float* Wp0 = (const float*)d_in[11];
  const float* Wp1 = (const float*)d_in[12];
  const float* Wp2 = (const float*)d_in[13];
  float* out = (float*)d_out;

  float* xs0 = (float*)d_ws;                        // 640,000 f
  float* xs1 = xs0 + (size_t)N_NODES * 64;          // 1,920,000 f
  float* agg = xs1 + (size_t)N_NODES * 192;         // 13,440,000 f (L2-resident)

  const float s128 = 0.088388347648f * 0.0625f;     // 1/sqrt(128) / AVG_NEIGH
  const float s192 = 0.072168783649f * 0.0625f;     // 1/sqrt(192) / AVG_NEIGH

  const int agg4 = N_NODES * 1344 / 4;
  zero_kernel4<<<(agg4 + 255) / 256, 256, 0, stream>>>((v4f*)agg, agg4);
  node_lin_t<false><<<dim3(N_NODES / 16, 1), 128, 0, stream>>>(nf, Wl0, xs0);
  node_lin_t<true ><<<dim3(N_NODES / 16, 3), 128, 0, stream>>>(nf, Wl1, xs1);
  edge_kernel<<<dim3(N_EDGES / 64), 128, 0, stream>>>(ea, ef, ei, Wm1, Wm2, Wm3,
                                                      Wm4, xs0, xs1, agg);
  out_lin_t<1, 2><<<dim3(N_NODES / 16, 1), 128, 0, stream>>>(agg, Wp0, out, 0,
                                                             0, 768, 0, s128);
  out_lin_t<3, 3><<<dim3(N_NODES / 16, 3), 128, 0, stream>>>(agg, Wp1, out, 1,
                                                             64, 576, 1152, s192);
  out_lin_t<5, 2><<<dim3(N_NODES / 16, 5), 128, 0, stream>>>(agg, Wp2, out, 4,
                                                             256, 832, 0, s128);

  (void)in_sizes; (void)n_in; (void)out_size; (void)ws_size;
}